// Loss_75711683494038
// MI455X (gfx1250) — compile-verified
//
#include <hip/hip_runtime.h>
#include <hip/hip_bf16.h>
#include <math.h>

// ---------------------------------------------------------------------------
// Problem constants (match reference)
// ---------------------------------------------------------------------------
#define BATCH    4096
#define FEATD    512
#define CLASSN   128
#define NROWS    8192          // 2*BATCH
#define QROWS    256           // 2*CLASSN
#define QCOLS    4096          // BATCH
#define INV_TF   2.0f          // 1/TEMP_F
#define COS_EPS  1e-8f

#define M_INIT   (-3.0e38f)    // online-softmax init (finite: no NaN paths)
#define MASKED   (-1.0e38f)    // masked logit sentinel: exp underflows to 0

typedef __attribute__((ext_vector_type(2))) float v2f;
typedef __attribute__((ext_vector_type(8))) float v8f;

// Fragment layout: per k-step of 4, 32 lanes x 2 floats, padded stride 66
// (even stride => 8B-aligned v2f, bank-conflict-free ds_load_b64 on the
// compute side and conflict-free staging stores).
#define FRAG_STRIDE 66

// Workspace layout (floats)
#define WS_FEAT   0            // 512 feature partials (128 WGs * 4 waves)
#define WS_LABEL  512          // 16 label partials
#define WS_NORM   528          // 256 row norms of q
#define WS_COLS   784          // 256 column sums (p_i | p_j, unnormalized)
#define WS_QT     2048         // qT: 256 x 4096 row-major  (needs ~4.3MB ws)

__device__ __forceinline__ v8f wmma_f32(v2f a, v2f b, v8f c) {
    // D = A(16x4,f32) * B(4x16,f32) + C(16x16,f32)  — exact fp32
    return __builtin_amdgcn_wmma_f32_16x16x4_f32(
        false, a, false, b, (short)0, c, false, false);
}

// ---------------------------------------------------------------------------
// CDNA5 async global->LDS DMA (ASYNCcnt-tracked, no VGPR round trip)
// ---------------------------------------------------------------------------
__device__ __forceinline__ void async_b64(const float* gsrc, unsigned lds_byte) {
    asm volatile("global_load_async_to_lds_b64 %0, %1, off"
                 :: "v"(lds_byte), "v"(gsrc)
                 : "memory");
}
__device__ __forceinline__ void wait_async0() {
    asm volatile("s_wait_asynccnt 0x0" ::: "memory");
}

// Async staging of a 16 x (NC4*4) fp32 tile into WMMA fragment order.
// Each float4 becomes two async b64 copies (lane pair lo -> +0, hi -> +128B).
template <int NC4>
__device__ __forceinline__ void stage_frag_async(const float* __restrict__ src,
                                                 int rowStride,
                                                 float* __restrict__ frag,
                                                 int tid, int nthreads) {
    const unsigned fbase = (unsigned)(uintptr_t)frag;   // low 32 bits = LDS offset
    for (int id = tid; id < 16 * NC4; id += nthreads) {
        const int m = id / NC4;          // row within tile (0..15)
        const int s = id % NC4;          // k-step (4 floats per step)
        const float* g = src + m * rowStride + (s << 2);
        const unsigned d = fbase + (unsigned)((s * FRAG_STRIDE + (m << 1)) << 2);
        async_b64(g,     d);             // lanes 0-15 : K = 4s, 4s+1
        async_b64(g + 2, d + 128);       // lanes 16-31: K = 4s+2, 4s+3
    }
}

// Branch-free online softmax step.
__device__ __forceinline__ void online_update(float& M, float& S, float x) {
    const float mn = fmaxf(M, x);
    S = S * __expf(M - mn) + __expf(x - mn);
    M = mn;
}

// ---------------------------------------------------------------------------
// Kernel 1: qT[r, b] = q1[b, r] (r<128) | q2[b, r-128]   (coalesced writes)
// ---------------------------------------------------------------------------
__global__ void transpose_q_kernel(const float* __restrict__ q1,
                                   const float* __restrict__ q2,
                                   float* __restrict__ qT) {
    const int idx = blockIdx.x * 256 + threadIdx.x;   // < 256*4096
    const int r = idx >> 12;
    const int b = idx & 4095;
    const float v = (r < CLASSN) ? q1[b * CLASSN + r]
                                 : q2[b * CLASSN + (r - CLASSN)];
    qT[idx] = v;
}

// ---------------------------------------------------------------------------
// Kernel 2: per-row norm and sum of qT (one block per row)
// ---------------------------------------------------------------------------
__global__ void rowstat_kernel(const float* __restrict__ qT,
                               float* __restrict__ norms,
                               float* __restrict__ colsums) {
    const int r = blockIdx.x;
    const int tid = threadIdx.x;
    float s = 0.f, ss = 0.f;
    for (int b = tid; b < QCOLS; b += 256) {
        const float v = qT[r * QCOLS + b];
        s += v; ss += v * v;
    }
    #pragma unroll
    for (int off = 1; off < 32; off <<= 1) {
        s  += __shfl_xor(s,  off, 32);
        ss += __shfl_xor(ss, off, 32);
    }
    __shared__ float red[16];
    const int w = tid >> 5, lane = tid & 31;
    if (lane == 0) { red[w] = s; red[8 + w] = ss; }
    __syncthreads();
    if (tid == 0) {
        float S = 0.f, SS = 0.f;
        for (int i = 0; i < 8; ++i) { S += red[i]; SS += red[8 + i]; }
        norms[r]   = sqrtf(SS);
        colsums[r] = S;
    }
}

// ---------------------------------------------------------------------------
// Kernel 3: feature NT-Xent, fused Gram + online logsumexp.
// 128 WGs x 4 waves; wave owns one 16-row tile (A resident in LDS).
// B is double-buffered: async DMA of tile j+1 overlaps the 128-WMMA K loop.
// ---------------------------------------------------------------------------
#define F_TPW 4                       // waves (row tiles) per WG
#define F_KS  (FEATD / 4)             // 128 k-steps
#define F_JT  (NROWS / 16)            // 512 column tiles
#define F_TILEF (F_KS * FRAG_STRIDE)  // floats per fragment tile

__device__ __forceinline__ const float* feat_tile_src(const float* h1,
                                                      const float* h2, int tile) {
    const int gr0 = tile * 16;
    return (gr0 < BATCH) ? (h1 + gr0 * FEATD) : (h2 + (gr0 - BATCH) * FEATD);
}

__global__ void feature_kernel(const float* __restrict__ h1,
                               const float* __restrict__ h2,
                               const float* __restrict__ conf,
                               float* __restrict__ partials) {
    extern __shared__ float lds[];
    float* ldsA = lds;                       // F_TPW tiles
    float* ldsB = lds + F_TPW * F_TILEF;     // 2 tiles (double buffer)

    const int tid  = threadIdx.x;            // 0..127
    const int lane = tid & 31;
    // Wave-uniform values scalarized so hot-loop branches are SALU-only.
    const int w       = __builtin_amdgcn_readfirstlane(tid >> 5);
    const int iw      = blockIdx.x * F_TPW + w;       // row tile (0..511)
    const int rowbase = iw * 16;
    const int partner = (iw < 256) ? iw + 256 : iw - 256;
    const int nthr = F_TPW * 32;

    // Stage the 4 A row tiles + B buffer 0 via async DMA, then one wait+barrier.
    for (int t = 0; t < F_TPW; ++t)
        stage_frag_async<F_KS>(feat_tile_src(h1, h2, blockIdx.x * F_TPW + t),
                               FEATD, ldsA + t * F_TILEF, tid, nthr);
    stage_frag_async<F_KS>(feat_tile_src(h1, h2, 0), FEATD, ldsB, tid, nthr);
    wait_async0();
    __syncthreads();

    float m_run[8], s_run[8], pos[8];
    #pragma unroll
    for (int v = 0; v < 8; ++v) { m_run[v] = M_INIT; s_run[v] = 0.f; pos[v] = 0.f; }

    const float* wA = ldsA + w * F_TILEF;

    for (int j = 0; j < F_JT; ++j) {
        // Kick off async staging of tile j+1 into the other buffer.
        if (j + 1 < F_JT)
            stage_frag_async<F_KS>(feat_tile_src(h1, h2, j + 1), FEATD,
                                   ldsB + ((j + 1) & 1) * F_TILEF, tid, nthr);

        const float* curB = ldsB + (j & 1) * F_TILEF;
        v8f acc = {0.f, 0.f, 0.f, 0.f, 0.f, 0.f, 0.f, 0.f};
        #pragma unroll 8
        for (int s = 0; s < F_KS; ++s) {
            const v2f a = *(const v2f*)(wA   + s * FRAG_STRIDE + (lane << 1));
            const v2f b = *(const v2f*)(curB + s * FRAG_STRIDE + (lane << 1));
            acc = wmma_f32(a, b, acc);
        }

        if (j != iw && j != partner) {
            // Hot path (510/512 tiles): scalar branch, no exec manipulation.
            #pragma unroll
            for (int v = 0; v < 8; ++v)
                online_update(m_run[v], s_run[v], acc[v] * INV_TF);
        } else {
            const bool jd = (j == iw), jp = (j == partner);
            #pragma unroll
            for (int v = 0; v < 8; ++v) {
                float x = acc[v] * INV_TF;
                const bool isD = (lane == v) || (lane == v + 24);
                if (jp && isD) pos[v] = x;
                if (jd && isD) x = MASKED;          // exp underflows to 0
                online_update(m_run[v], s_run[v], x);
            }
        }

        wait_async0();      // own async copies for buffer (j+1)&1 done
        __syncthreads();    // all waves' copies done / buffer j&1 free
    }

    // Combine per-row (max, sumexp) across the 16 lanes holding that row.
    float lsum = 0.f;
    #pragma unroll
    for (int v = 0; v < 8; ++v) {
        float M = m_run[v], S = s_run[v];
        #pragma unroll
        for (int off = 1; off < 16; off <<= 1) {
            const float M2 = __shfl_xor(M, off, 32);
            const float S2 = __shfl_xor(S, off, 32);
            const float mn = fmaxf(M, M2);
            S = S * __expf(M - mn) + S2 * __expf(M2 - mn);
            M = mn;
        }
        const float lse = M + __logf(S);
        const bool isD = (lane == v) || (lane == v + 24);
        if (isD) {
            const int R = rowbase + v + ((lane >> 4) << 3);
            lsum += (lse - pos[v]) * conf[R & (BATCH - 1)];
        }
    }
    #pragma unroll
    for (int off = 1; off < 32; off <<= 1) lsum += __shfl_xor(lsum, off, 32);
    if (lane == 0) partials[blockIdx.x * F_TPW + w] = lsum;
}

// ---------------------------------------------------------------------------
// Kernel 4: label NT-Xent (cosine sim over qT, K=4096 chunked by 256).
// 2 WGs x 8 waves; wave owns one 16-row tile of the 256-row q matrix.
// ---------------------------------------------------------------------------
#define L_KC   256                    // k chunk
#define L_NKC  (QCOLS / L_KC)         // 16 chunks
#define L_ST   (L_KC / 4)             // 64 steps per chunk
#define L_TILEF (L_ST * FRAG_STRIDE)

__global__ void label_kernel(const float* __restrict__ qT,
                             const float* __restrict__ norms,
                             float* __restrict__ partials) {
    extern __shared__ float lds[];
    float* ldsA = lds;                   // 8 tiles (one chunk each)
    float* ldsB = lds + 8 * L_TILEF;

    const int tid  = threadIdx.x;        // 0..255
    const int lane = tid & 31;
    const int w       = __builtin_amdgcn_readfirstlane(tid >> 5);
    const int iw      = blockIdx.x * 8 + w;  // row tile 0..15
    const int rowbase = iw * 16;
    const int partner = (iw < 8) ? iw + 8 : iw - 8;

    float nr[8];
    #pragma unroll
    for (int v = 0; v < 8; ++v)
        nr[v] = norms[rowbase + v + ((lane >> 4) << 3)];

    float m_run[8], s_run[8], pos[8];
    #pragma unroll
    for (int v = 0; v < 8; ++v) { m_run[v] = M_INIT; s_run[v] = 0.f; pos[v] = 0.f; }

    const float* wA = ldsA + w * L_TILEF;

    for (int j = 0; j < 16; ++j) {
        v8f acc = {0.f, 0.f, 0.f, 0.f, 0.f, 0.f, 0.f, 0.f};
        for (int kc = 0; kc < L_NKC; ++kc) {
            __syncthreads();             // previous chunk fully consumed
            for (int t = 0; t < 8; ++t)
                stage_frag_async<L_ST>(
                    qT + ((blockIdx.x * 8 + t) * 16) * QCOLS + kc * L_KC,
                    QCOLS, ldsA + t * L_TILEF, tid, 256);
            stage_frag_async<L_ST>(qT + (j * 16) * QCOLS + kc * L_KC, QCOLS,
                                   ldsB, tid, 256);
            wait_async0();
            __syncthreads();
            #pragma unroll 8
            for (int s = 0; s < L_ST; ++s) {
                const v2f a = *(const v2f*)(wA   + s * FRAG_STRIDE + (lane << 1));
                const v2f b = *(const v2f*)(ldsB + s * FRAG_STRIDE + (lane << 1));
                acc = wmma_f32(a, b, acc);
            }
        }
        const float nc = norms[j * 16 + (lane & 15)];
        if (j != iw && j != partner) {
            #pragma unroll
            for (int v = 0; v < 8; ++v)
                online_update(m_run[v], s_run[v],
                              acc[v] / fmaxf(nr[v] * nc, COS_EPS)); // TEMP_L=1
        } else {
            const bool jd = (j == iw), jp = (j == partner);
            #pragma unroll
            for (int v = 0; v < 8; ++v) {
                float x = acc[v] / fmaxf(nr[v] * nc, COS_EPS);
                const bool isD = (lane == v) || (lane == v + 24);
                if (jp && isD) pos[v] = x;
                if (jd && isD) x = MASKED;
                online_update(m_run[v], s_run[v], x);
            }
        }
    }

    float lsum = 0.f;
    #pragma unroll
    for (int v = 0; v < 8; ++v) {
        float M = m_run[v], S = s_run[v];
        #pragma unroll
        for (int off = 1; off < 16; off <<= 1) {
            const float M2 = __shfl_xor(M, off, 32);
            const float S2 = __shfl_xor(S, off, 32);
            const float mn = fmaxf(M, M2);
            S = S * __expf(M - mn) + S2 * __expf(M2 - mn);
            M = mn;
        }
        const float lse = M + __logf(S);
        const bool isD = (lane == v) || (lane == v + 24);
        if (isD) lsum += (lse - pos[v]);
    }
    #pragma unroll
    for (int off = 1; off < 32; off <<= 1) lsum += __shfl_xor(lsum, off, 32);
    if (lane == 0) partials[iw] = lsum;
}

// ---------------------------------------------------------------------------
// Kernel 5: finalize (fixed-order sums -> deterministic)
// ---------------------------------------------------------------------------
__global__ void finalize_kernel(const float* __restrict__ ws,
                                float* __restrict__ out) {
    const int lane = threadIdx.x;  // 32 threads
    float f = 0.f;
    for (int i = lane; i < 512; i += 32) f += ws[WS_FEAT + i];
    float l = (lane < 16) ? ws[WS_LABEL + lane] : 0.f;
    #pragma unroll
    for (int off = 1; off < 32; off <<= 1) {
        f += __shfl_xor(f, off, 32);
        l += __shfl_xor(l, off, 32);
    }
    if (lane == 0) {
        const float* cs = ws + WS_COLS;
        float si = 0.f, sj = 0.f;
        for (int c = 0; c < CLASSN; ++c) { si += cs[c]; sj += cs[CLASSN + c]; }
        float nei = __logf((float)CLASSN), nej = nei;
        for (int c = 0; c < CLASSN; ++c) {
            const float p = cs[c] / si;
            const float q = cs[CLASSN + c] / sj;
            nei += p * __logf(p);
            nej += q * __logf(q);
        }
        const float feat_mean  = f / (float)NROWS;
        const float label_mean = l / (float)QROWS;
        out[0] = 0.5f * feat_mean + 0.5f * (label_mean + nei + nej);
    }
}

// ---------------------------------------------------------------------------
extern "C" void kernel_launch(void* const* d_in, const int* in_sizes, int n_in,
                              void* d_out, int out_size, void* d_ws,
                              size_t ws_size, hipStream_t stream) {
    const float* h1   = (const float*)d_in[0];
    const float* h2   = (const float*)d_in[1];
    const float* q1   = (const float*)d_in[2];
    const float* q2   = (const float*)d_in[3];
    const float* conf = (const float*)d_in[4];

    float* ws   = (float*)d_ws;
    float* fpar = ws + WS_FEAT;
    float* lpar = ws + WS_LABEL;
    float* nrm  = ws + WS_NORM;
    float* cols = ws + WS_COLS;
    float* qT   = ws + WS_QT;

    transpose_q_kernel<<<(QROWS * QCOLS) / 256, 256, 0, stream>>>(q1, q2, qT);
    rowstat_kernel<<<QROWS, 256, 0, stream>>>(qT, nrm, cols);

    const size_t featLds = (size_t)(F_TPW + 2) * F_TILEF * sizeof(float); // ~198KB
    feature_kernel<<<NROWS / (16 * F_TPW), F_TPW * 32, featLds, stream>>>(
        h1, h2, conf, fpar);

    const size_t labLds = (size_t)(8 + 1) * L_TILEF * sizeof(float);      // ~149KB
    label_kernel<<<2, 256, labLds, stream>>>(qT, nrm, lpar);

    finalize_kernel<<<1, 32, 0, stream>>>(ws, (float*)d_out);
}